// QuonvLayer_9268539425285
// MI455X (gfx1250) — compile-verified
//
#include <hip/hip_runtime.h>
#include <math.h>

// ---------------------------------------------------------------------------
// Quanvolution layer, restructured as:
//   (1) build 16x16 circuit matrix U from weights (tiny kernel, workspace)
//   (2) per patch: product-state embed -> Y = U * S0 via V_WMMA_F32_16X16X4_F32
//       (4 chained K-blocks, f32 accumulate) -> signed |y|^2 reductions
// One wave32 handles 16 patches (one WMMA N-tile).
// ---------------------------------------------------------------------------

typedef __attribute__((ext_vector_type(2))) float v2f;
typedef __attribute__((ext_vector_type(8))) float v8f;

#define BIMG 64
#define HIMG 128
#define WIMG 128
#define HO   127
#define WO   127
#define NPATCH (BIMG * HO * WO)   // 1,032,256 (multiple of 16)

// ---------------------------------------------------------------------------
// Build U[16][16] (row-major): column j = circuit applied to basis state |j>.
// Wire w occupies bit (3-w) of the flattened index (C-order of (2,2,2,2)).
// Layers: for l in {0,1}: RY(weights[l][w]) on each wire, then CNOT ring.
// ---------------------------------------------------------------------------
__global__ void build_u_kernel(const float* __restrict__ w, float* __restrict__ U) {
    int j = threadIdx.x;
    if (j >= 16) return;
    float st[16];
#pragma unroll
    for (int i = 0; i < 16; ++i) st[i] = (i == j) ? 1.0f : 0.0f;

#pragma unroll
    for (int l = 0; l < 2; ++l) {
        // RY(weights[l][q]) on each wire
#pragma unroll
        for (int q = 0; q < 4; ++q) {
            float s, c;
            __sincosf(0.5f * w[l * 4 + q], &s, &c);
            const int bit = 1 << (3 - q);
#pragma unroll
            for (int i = 0; i < 16; ++i) {
                if (!(i & bit)) {
                    float a = st[i], b = st[i | bit];
                    st[i]       = c * a - s * b;
                    st[i | bit] = s * a + c * b;
                }
            }
        }
        // CNOT(q, (q+1)%4): for ctrl bit set, flip target bit (permutation)
#pragma unroll
        for (int q = 0; q < 4; ++q) {
            const int cb = 1 << (3 - q);
            const int tb = 1 << (3 - ((q + 1) & 3));
            float tmp[16];
#pragma unroll
            for (int i = 0; i < 16; ++i) tmp[i] = st[(i & cb) ? (i ^ tb) : i];
#pragma unroll
            for (int i = 0; i < 16; ++i) st[i] = tmp[i];
        }
    }
#pragma unroll
    for (int i = 0; i < 16; ++i) U[i * 16 + j] = st[i];
}

// ---------------------------------------------------------------------------
// Main kernel: one wave = 16 patches.
// ---------------------------------------------------------------------------
__global__ __launch_bounds__(256) void quonv_wmma_kernel(
    const float* __restrict__ img, const float* __restrict__ U,
    float* __restrict__ out) {

    const int lane = threadIdx.x & 31;
    const bool lo  = lane < 16;
    const int  p   = lane & 15;                       // patch-in-wave == U row M
    const unsigned wave = (blockIdx.x * blockDim.x + threadIdx.x) >> 5;
    const unsigned q0 = wave * 16u;
    if (q0 >= NPATCH) return;                         // wave-uniform exit

    unsigned q  = q0 + (unsigned)p;
    unsigned qc = q < NPATCH ? q : (NPATCH - 1);      // clamp (keep EXEC full)

    // decode (b, i, j) from flat patch index
    const unsigned hw = HO * WO;
    unsigned b   = qc / hw;
    unsigned rem = qc - b * hw;
    unsigned ii  = rem / WO;
    unsigned jj  = rem - ii * WO;

    const float* base = img + ((b * HIMG + ii) * WIMG + jj);
    float p0 = base[0];
    float p1 = base[1];
    float p2 = base[WIMG];
    float p3 = base[WIMG + 1];

    // Angle embedding: half-angles pi*p/2
    const float HPI = 1.57079632679489662f;
    float c0, s0, c1, s1, c2, s2, c3, s3;
    __sincosf(HPI * p0, &s0, &c0);
    __sincosf(HPI * p1, &s1, &c1);
    __sincosf(HPI * p2, &s2, &c2);
    __sincosf(HPI * p3, &s3, &c3);

    // s0 amplitude(i) = f0(b0)f1(b1)f2(b2)f3(b3), i = b0*8+b1*4+b2*2+b3.
    // B-matrix lane split: low lanes carry rows with (b2,b3) in {(0,0),(0,1)},
    // high lanes {(1,0),(1,1)}; k-block kb selects (b0,b1).
    float g0 = c0 * c1, g1 = c0 * s1, g2 = s0 * c1, g3 = s0 * s1;
    float h0 = lo ? (c2 * c3) : (s2 * c3);
    float h1 = lo ? (c2 * s3) : (s2 * s3);
    v2f b0v = {g0 * h0, g0 * h1};
    v2f b1v = {g1 * h0, g1 * h1};
    v2f b2v = {g2 * h0, g2 * h1};
    v2f b3v = {g3 * h0, g3 * h1};

    // A-matrix: U row M = p, K-slice per block; low lanes K 0/1, high K 2/3.
    const float4* Urow = (const float4*)(U + p * 16);
    float4 u0 = Urow[0], u1 = Urow[1], u2 = Urow[2], u3 = Urow[3];
    v2f a0 = lo ? (v2f){u0.x, u0.y} : (v2f){u0.z, u0.w};
    v2f a1 = lo ? (v2f){u1.x, u1.y} : (v2f){u1.z, u1.w};
    v2f a2 = lo ? (v2f){u2.x, u2.y} : (v2f){u2.z, u2.w};
    v2f a3 = lo ? (v2f){u3.x, u3.y} : (v2f){u3.z, u3.w};

    // Y[16 x 16patches] = U * S0, chained over K in blocks of 4
    v8f acc = {};
    acc = __builtin_amdgcn_wmma_f32_16x16x4_f32(false, a0, false, b0v, (short)0, acc, false, false);
    acc = __builtin_amdgcn_wmma_f32_16x16x4_f32(false, a1, false, b1v, (short)0, acc, false, false);
    acc = __builtin_amdgcn_wmma_f32_16x16x4_f32(false, a2, false, b2v, (short)0, acc, false, false);
    acc = __builtin_amdgcn_wmma_f32_16x16x4_f32(false, a3, false, b3v, (short)0, acc, false, false);

    // Signed probability sums. Lane L<16 holds rows 0..7, L+16 rows 8..15.
    float sq[8];
#pragma unroll
    for (int r = 0; r < 8; ++r) sq[r] = acc[r] * acc[r];

    float sAll  = ((sq[0] + sq[1]) + (sq[2] + sq[3])) + ((sq[4] + sq[5]) + (sq[6] + sq[7]));
    float part0 = lo ? sAll : -sAll;                                    // bit3
    float part1 = ((sq[0] + sq[1]) + (sq[2] + sq[3])) - ((sq[4] + sq[5]) + (sq[6] + sq[7])); // bit2
    float part2 = ((sq[0] + sq[1]) + (sq[4] + sq[5])) - ((sq[2] + sq[3]) + (sq[6] + sq[7])); // bit1
    float part3 = ((sq[0] + sq[2]) + (sq[4] + sq[6])) - ((sq[1] + sq[3]) + (sq[5] + sq[7])); // bit0

    float z0 = part0 + __shfl_xor(part0, 16, 32);
    float z1 = part1 + __shfl_xor(part1, 16, 32);
    float z2 = part2 + __shfl_xor(part2, 16, 32);
    float z3 = part3 + __shfl_xor(part3, 16, 32);

    if (q < NPATCH) {
        // out[q][0..3]; low lanes write (z0,z1), high lanes (z2,z3) ->
        // wave writes 16 contiguous float4s (256B coalesced).
        float2* o = (float2*)(out + (size_t)q * 4 + (lo ? 0 : 2));
        *o = lo ? make_float2(z0, z1) : make_float2(z2, z3);
    }
}

// ---------------------------------------------------------------------------
extern "C" void kernel_launch(void* const* d_in, const int* in_sizes, int n_in,
                              void* d_out, int out_size, void* d_ws, size_t ws_size,
                              hipStream_t stream) {
    const float* img     = (const float*)d_in[0];  // [64,128,128,1] f32
    const float* weights = (const float*)d_in[1];  // [2,4] f32
    float*       out     = (float*)d_out;          // [64,127,127,4] f32
    float*       Uws     = (float*)d_ws;           // 16*16 floats scratch

    build_u_kernel<<<1, 32, 0, stream>>>(weights, Uws);

    const int waves  = (NPATCH + 15) / 16;         // 64,516
    const int wpb    = 256 / 32;                   // 8 waves / block
    const int blocks = (waves + wpb - 1) / wpb;    // 8,065
    quonv_wmma_kernel<<<blocks, 256, 0, stream>>>(img, Uws, out);
}